// RTDETRPostProcessor_43628277793040
// MI455X (gfx1250) — compile-verified
//
#include <hip/hip_runtime.h>
#include <cstdint>
#include <cstddef>
#include <math.h>

#define NUM_CLASSES 80
#define NQ 300
#define TOPK 300
#define NELEM (NQ * NUM_CLASSES)   // 24000 candidates per image
#define NVEC  (NELEM / 4)          // 6000 uint4 chunks
#define THREADS 256
#define NWAVE (THREADS / 32)       // 8 waves (wave32)
#define CAP 512                    // candidate list capacity (>= TOPK + tie slack)

// Dynamic LDS layout (bytes) -- NO static __shared__ so this region starts at
// LDS offset 0 and keys are 16B aligned for async b128 writes:
//   [0,      96000)  keys : NELEM u32 (raw logit bits, then sortable keys)
//   [96000, 104192)  hist : 8 wave-private copies x 256 u32
//   [104192,108288)  cand : 512 u64  (104192 % 8 == 0)
//   [108288,108304)  ctrl : prefix, remaining, nc, pad
#define KEYS_OFF  0
#define HIST_OFF  96000
#define CAND_OFF  104192
#define CTRL_OFF  108288
#define LDS_BYTES 108304

// float bits -> order-preserving unsigned key (bigger key == bigger float)
__device__ __forceinline__ uint32_t f2k(uint32_t b) {
  return (b & 0x80000000u) ? ~b : (b ^ 0x80000000u);
}
// inverse: key -> float
__device__ __forceinline__ float k2f(uint32_t u) {
  uint32_t b = (u & 0x80000000u) ? (u ^ 0x80000000u) : ~u;
  float f;
  __builtin_memcpy(&f, &b, 4);
  return f;
}

__global__ void __launch_bounds__(THREADS)
rtdetr_post_kernel(const float* __restrict__ logits,
                   const float* __restrict__ boxes,
                   const float* __restrict__ osz,
                   float* __restrict__ out, int B) {
  extern __shared__ __align__(16) unsigned char smem[];
  uint32_t*           keys  = (uint32_t*)(smem + KEYS_OFF);
  uint4*              keys4 = (uint4*)(smem + KEYS_OFF);
  unsigned int*       hist  = (unsigned int*)(smem + HIST_OFF);   // 8 x 256
  unsigned long long* cand  = (unsigned long long*)(smem + CAND_OFF);
  unsigned int*       ctrl  = (unsigned int*)(smem + CTRL_OFF);
  // ctrl[0] = prefix, ctrl[1] = remaining, ctrl[2] = nc

  const int b   = blockIdx.x;
  const int tid = threadIdx.x;
  const int wid = tid >> 5;          // wave id (wave32)
  unsigned int* myhist = hist + wid * 256;
  const float* lg = logits + (size_t)b * NELEM;

  // Prefetch the box row we will gather from later (global_prefetch_b8).
  {
    const char* pb = (const char*)(boxes + (size_t)b * NQ * 4);
    for (int off = tid * 128; off < NQ * 16; off += THREADS * 128)
      __builtin_prefetch(pb + off, 0, 1);
  }

  // ---- Stage logits to LDS via CDNA5 async DMA (ASYNCcnt-tracked) ----
  // 24000 floats = 6000 x 16B chunks; each lane supplies its own LDS offset
  // (low 32 bits of the generic shared pointer == allocation-relative offset,
  //  16B aligned since keys sits at dynamic-LDS offset 0).
  {
    const uint32_t lds_base = (uint32_t)(size_t)(void*)keys;
    const uint64_t g_base   = (uint64_t)(size_t)(const void*)lg;
    for (uint32_t c = tid; c < (uint32_t)NVEC; c += THREADS) {
      uint32_t l = lds_base + c * 16u;
      uint64_t g = g_base + (uint64_t)c * 16u;
      asm volatile("global_load_async_to_lds_b128 %0, %1, off"
                   :: "v"(l), "v"(g) : "memory");
    }
  }
  // Overlap LDS setup with the in-flight DMA (these regions are not DMA targets)
  for (int i = tid; i < NWAVE * 256; i += THREADS) hist[i] = 0u;
  if (tid == 0) { ctrl[0] = 0u; ctrl[1] = (unsigned int)TOPK; ctrl[2] = 0u; }
  asm volatile("s_wait_asynccnt 0" ::: "memory");
  __syncthreads();

  // ---- Round 1 (top byte): fused transform + histogram, b128 LDS reads ----
  for (int i = tid; i < NVEC; i += THREADS) {
    uint4 v = keys4[i];
    v.x = f2k(v.x); v.y = f2k(v.y); v.z = f2k(v.z); v.w = f2k(v.w);
    keys4[i] = v;
    atomicAdd(&myhist[v.x >> 24], 1u);
    atomicAdd(&myhist[v.y >> 24], 1u);
    atomicAdd(&myhist[v.z >> 24], 1u);
    atomicAdd(&myhist[v.w >> 24], 1u);
  }
  __syncthreads();
  {
    unsigned int s = 0;
    for (int w = 0; w < NWAVE; ++w) s += hist[w * 256 + tid];
    hist[tid] = s;
  }
  __syncthreads();
  if (tid == 0) {
    int rem = (int)ctrl[1];
    unsigned int acc = 0;
    int d = 0;
    for (int bin = 255; bin >= 0; --bin) {
      unsigned int c = hist[bin];
      if (acc + c >= (unsigned int)rem) { d = bin; break; }
      acc += c;
    }
    ctrl[0] = (uint32_t)d << 24;
    ctrl[1] = (unsigned int)(rem - (int)acc);
  }
  __syncthreads();

  // ---- Rounds 2..4: filtered histograms over fixed prefix ----
  for (int shift = 16; shift >= 0; shift -= 8) {
    for (int i = tid; i < NWAVE * 256; i += THREADS) hist[i] = 0u;
    __syncthreads();
    const uint32_t pref_hi = ctrl[0] >> (shift + 8);
    for (int i = tid; i < NVEC; i += THREADS) {
      uint4 v = keys4[i];
      if ((v.x >> (shift + 8)) == pref_hi) atomicAdd(&myhist[(v.x >> shift) & 0xFFu], 1u);
      if ((v.y >> (shift + 8)) == pref_hi) atomicAdd(&myhist[(v.y >> shift) & 0xFFu], 1u);
      if ((v.z >> (shift + 8)) == pref_hi) atomicAdd(&myhist[(v.z >> shift) & 0xFFu], 1u);
      if ((v.w >> (shift + 8)) == pref_hi) atomicAdd(&myhist[(v.w >> shift) & 0xFFu], 1u);
    }
    __syncthreads();
    {
      unsigned int s = 0;
      for (int w = 0; w < NWAVE; ++w) s += hist[w * 256 + tid];
      hist[tid] = s;
    }
    __syncthreads();
    if (tid == 0) {
      int rem = (int)ctrl[1];
      unsigned int acc = 0;
      int d = 0;
      for (int bin = 255; bin >= 0; --bin) {
        unsigned int c = hist[bin];
        if (acc + c >= (unsigned int)rem) { d = bin; break; }
        acc += c;
      }
      ctrl[0] |= ((uint32_t)d << shift);
      ctrl[1] = (unsigned int)(rem - (int)acc);
    }
    __syncthreads();
  }
  const uint32_t T = ctrl[0];  // key of the K-th largest element

  // ---- Collect all candidates with key >= T (b128 LDS reads) ----
  for (int i = tid; i < CAP; i += THREADS) cand[i] = 0ull;
  __syncthreads();
  for (int i = tid; i < NVEC; i += THREADS) {
    uint4 v = keys4[i];
    uint32_t i0 = (uint32_t)i * 4u;
    #pragma unroll
    for (int lane = 0; lane < 4; ++lane) {
      uint32_t k = (lane == 0) ? v.x : (lane == 1) ? v.y : (lane == 2) ? v.z : v.w;
      if (k >= T) {
        unsigned int pos = atomicAdd(&ctrl[2], 1u);
        if (pos < CAP) {
          // composite sorts: descending key, then ascending original index
          cand[pos] = ((unsigned long long)k << 32) |
                      (unsigned long long)(0xFFFFFFFFu - (i0 + (uint32_t)lane));
        }
      }
    }
  }
  __syncthreads();

  // ---- Bitonic sort 512 composites, descending (stable tie order via ~idx) ----
  for (unsigned int kk = 2; kk <= CAP; kk <<= 1) {
    for (unsigned int j = kk >> 1; j > 0; j >>= 1) {
      __syncthreads();
      for (int w = 0; w < 2; ++w) {
        unsigned int i   = (unsigned int)tid + (unsigned int)w * THREADS;
        unsigned int ixj = i ^ j;
        if (ixj > i) {
          bool descBlock = ((i & kk) == 0);
          unsigned long long a = cand[i], c = cand[ixj];
          bool doswap = descBlock ? (a < c) : (a > c);
          if (doswap) { cand[i] = c; cand[ixj] = a; }
        }
      }
    }
  }
  __syncthreads();

  // ---- Decode + emit top-300 ----
  const float W = osz[b * 2 + 0];
  const float H = osz[b * 2 + 1];
  float* out_labels = out;
  float* out_boxes  = out + (size_t)B * TOPK;
  float* out_scores = out + (size_t)B * TOPK * 5;
  for (int r = tid; r < TOPK; r += THREADS) {
    unsigned long long c = cand[r];
    uint32_t idx = 0xFFFFFFFFu - (uint32_t)(c & 0xFFFFFFFFull);
    uint32_t u   = (uint32_t)(c >> 32);
    float logit  = k2f(u);
    float score  = 1.0f / (1.0f + expf(-logit));
    uint32_t cls = idx % NUM_CLASSES;
    uint32_t q   = idx / NUM_CLASSES;

    const float4 bb = ((const float4*)boxes)[(size_t)b * NQ + q];
    float hw = 0.5f * bb.z, hh = 0.5f * bb.w;
    float4 o;
    o.x = (bb.x - hw) * W;
    o.y = (bb.y - hh) * H;
    o.z = (bb.x + hw) * W;
    o.w = (bb.y + hh) * H;

    size_t base = (size_t)b * TOPK + (size_t)r;
    out_labels[base] = (float)cls;
    ((float4*)out_boxes)[base] = o;
    out_scores[base] = score;
  }
}

extern "C" void kernel_launch(void* const* d_in, const int* in_sizes, int n_in,
                              void* d_out, int out_size, void* d_ws, size_t ws_size,
                              hipStream_t stream) {
  (void)n_in; (void)out_size; (void)d_ws; (void)ws_size;
  const float* logits = (const float*)d_in[0];
  const float* boxes  = (const float*)d_in[1];
  const float* osz    = (const float*)d_in[2];
  int B = in_sizes[2] / 2;  // orig_target_sizes is [B,2]
  dim3 grid((unsigned int)B), block(THREADS);
  hipLaunchKernelGGL(rtdetr_post_kernel, grid, block, LDS_BYTES, stream,
                     logits, boxes, osz, (float*)d_out, B);
}